// VectorQuantizer_21311627723517
// MI455X (gfx1250) — compile-verified
//
#include <hip/hip_runtime.h>
#include <hip/hip_bf16.h>
#include <math.h>

typedef __attribute__((ext_vector_type(2))) float v2f;
typedef __attribute__((ext_vector_type(8))) float v8f;

#define D_DIM 64
#define K_DIM 2048
#define N_ROWS 32768
#define NCHUNK 16
#define NCHUNKS (K_DIM / NCHUNK)                         // 128
#define ROWS_PER_WAVE 32                                 // two 16-row C tiles
#define WAVES_PER_BLOCK 8
#define ROWS_PER_BLOCK (ROWS_PER_WAVE * WAVES_PER_BLOCK) // 256
#define MAIN_BLOCKS (N_ROWS / ROWS_PER_BLOCK)            // 128
#define N_LOSS_PARTS (MAIN_BLOCKS * WAVES_PER_BLOCK)     // 1024

// ---------------------------------------------------------------------------
// Kernel 1: per-code squared norms ||w_k||^2, zero histogram.
// w is [D,K] row-major, so thread k reads w[d*K + k] -> fully coalesced.
// ---------------------------------------------------------------------------
__global__ void vq_init_kernel(const float* __restrict__ w,
                               float* __restrict__ wsq,
                               unsigned int* __restrict__ counts) {
    int k = blockIdx.x * blockDim.x + threadIdx.x;
    if (k < K_DIM) {
        float s = 0.0f;
        #pragma unroll 8
        for (int d = 0; d < D_DIM; ++d) {
            float v = w[d * K_DIM + k];
            s += v * v;
        }
        wsq[k] = s;
        counts[k] = 0u;
    }
}

// ---------------------------------------------------------------------------
// Kernel 2: main VQ kernel.
// Block = 256 threads = 8 waves; each wave owns TWO 16-row tiles (32 rows).
// Double-buffered LDS staging of 64x16 w-chunks (one barrier per chunk,
// global->LDS fetch overlapped with WMMA). Each B fragment feeds two
// V_WMMA_F32_16X16X4_F32 ops (c0/c1), halving DScnt waits per WMMA and
// halving per-block w traffic from L2.
// ---------------------------------------------------------------------------
__global__ void __launch_bounds__(256)
vq_main_kernel(const float* __restrict__ x,      // [N_ROWS, D]
               const float* __restrict__ w,      // [D, K]
               const float* __restrict__ wsq,    // [K]
               float* __restrict__ q_out,        // [N_ROWS, D]
               unsigned int* __restrict__ counts,// [K]
               float* __restrict__ loss_parts)   // [N_LOSS_PARTS]
{
    __shared__ float wtile[2][D_DIM * NCHUNK];   // 2 x 4 KB
    __shared__ int   idx_sh[ROWS_PER_BLOCK];     // 256 ints

    const int lane = threadIdx.x & 31;
    const int wave = threadIdx.x >> 5;
    const int half = lane >> 4;   // 0: lanes 0-15, 1: lanes 16-31
    const int l16  = lane & 15;
    const int row0 = blockIdx.x * ROWS_PER_BLOCK + wave * ROWS_PER_WAVE;

    // A fragments for two 16x64 row tiles (16 k-steps of 16x4 fp32 each).
    // ISA layout: lanes 0-15 hold M=lane {K=0 in v0, K=1 in v1};
    // lanes 16-31 hold M=lane-16 {K=2 in v0, K=3 in v1}.
    v2f afragA[16], afragB[16];
    {
        const float* xrowA = x + (size_t)(row0 + l16) * D_DIM;
        const float* xrowB = xrowA + 16 * D_DIM;
        #pragma unroll
        for (int s = 0; s < 16; ++s) {
            int d = 4 * s + 2 * half;
            afragA[s].x = xrowA[d];
            afragA[s].y = xrowA[d + 1];
            afragB[s].x = xrowB[d];
            afragB[s].y = xrowB[d + 1];
        }
    }

    float best0[8], best1[8];
    int   bidx0[8], bidx1[8];
    #pragma unroll
    for (int v = 0; v < 8; ++v) {
        best0[v] = 3.4e38f; bidx0[v] = 0;
        best1[v] = 3.4e38f; bidx1[v] = 0;
    }

    // Stage chunk 0 into buffer 0.
    for (int t = threadIdx.x; t < D_DIM * NCHUNK; t += 256) {
        int d = t >> 4, n = t & 15;
        wtile[0][t] = w[d * K_DIM + n];
    }
    __syncthreads();

    for (int ci = 0; ci < NCHUNKS; ++ci) {
        const int k0 = ci * NCHUNK;

        // Stage next chunk into the other buffer; global loads issue now and
        // overlap with the WMMA chain below.
        if (ci + 1 < NCHUNKS) {
            float* nxt = wtile[(ci + 1) & 1];
            for (int t = threadIdx.x; t < D_DIM * NCHUNK; t += 256) {
                int d = t >> 4, n = t & 15;
                nxt[t] = w[d * K_DIM + k0 + NCHUNK + n];
            }
            // Prefetch chunk+2 into the cache hierarchy (global_prefetch_b8).
            if (ci + 2 < NCHUNKS && (threadIdx.x & 3) == 0)
                __builtin_prefetch(&w[(threadIdx.x >> 2) * K_DIM + k0 + 2 * NCHUNK], 0, 0);
        }

        // 16 fp32 WMMA k-steps; B fragment pipelined one step ahead and
        // shared by the two accumulators (c0: rows 0-15, c1: rows 16-31).
        const float* cur = wtile[ci & 1];
        v2f b_cur, b_nxt;
        {
            int kb = 2 * half;  // v0 = row K=(0|2), v1 = row K=(1|3)
            b_cur.x = cur[kb * NCHUNK + l16];
            b_cur.y = cur[(kb + 1) * NCHUNK + l16];
        }
        v8f c0 = {}, c1 = {};
        #pragma unroll
        for (int s = 0; s < 16; ++s) {
            if (s + 1 < 16) {
                int kb = 4 * (s + 1) + 2 * half;
                b_nxt.x = cur[kb * NCHUNK + l16];
                b_nxt.y = cur[(kb + 1) * NCHUNK + l16];
            }
            c0 = __builtin_amdgcn_wmma_f32_16x16x4_f32(
                    false, afragA[s], false, b_cur, (short)0, c0, false, false);
            c1 = __builtin_amdgcn_wmma_f32_16x16x4_f32(
                    false, afragB[s], false, b_cur, (short)0, c1, false, false);
            b_cur = b_nxt;
        }

        // Score = ||w_n||^2 - 2*dot (||x_m||^2 constant in n -> dropped).
        float wq = wsq[k0 + l16];
        #pragma unroll
        for (int v = 0; v < 8; ++v) {
            float s0 = wq - 2.0f * c0[v];
            if (s0 < best0[v]) { best0[v] = s0; bidx0[v] = k0 + l16; }
            float s1 = wq - 2.0f * c1[v];
            if (s1 < best1[v]) { best1[v] = s1; bidx1[v] = k0 + l16; }
        }

        // One barrier per chunk: orders this iter's LDS reads before the
        // buffer is overwritten, and next-buffer stores before next reads.
        __syncthreads();
    }

    // Argmin reduction across the 16 lanes of each half (ties -> lower index,
    // matching jnp.argmin). XOR offsets 8..1 stay within each 16-lane group.
    #pragma unroll
    for (int off = 8; off >= 1; off >>= 1) {
        #pragma unroll
        for (int v = 0; v < 8; ++v) {
            float ov; int oi;
            ov = __shfl_xor(best0[v], off, 32);
            oi = __shfl_xor(bidx0[v], off, 32);
            if (ov < best0[v] || (ov == best0[v] && oi < bidx0[v])) {
                best0[v] = ov; bidx0[v] = oi;
            }
            ov = __shfl_xor(best1[v], off, 32);
            oi = __shfl_xor(bidx1[v], off, 32);
            if (ov < best1[v] || (ov == best1[v] && oi < bidx1[v])) {
                best1[v] = ov; bidx1[v] = oi;
            }
        }
    }
    // C/D layout: VGPR v holds row m = v + 8*half. Lanes 0 & 16 publish.
    if (l16 == 0) {
        #pragma unroll
        for (int v = 0; v < 8; ++v) {
            idx_sh[wave * 32 + v + 8 * half]      = bidx0[v];
            idx_sh[wave * 32 + 16 + v + 8 * half] = bidx1[v];
        }
    }
    __syncthreads();

    // Histogram (integer atomics: deterministic). 32 rows, 32 lanes.
    atomicAdd(&counts[idx_sh[wave * 32 + lane]], 1u);

    // Gather codes, write quantized output, accumulate (q-x)^2.
    float lsum = 0.0f;
    #pragma unroll 4
    for (int it = 0; it < (ROWS_PER_WAVE * D_DIM) / 32; ++it) { // 64 iters
        int e = it * 32 + lane;
        int m = e >> 6;           // row within wave tile
        int d = e & 63;
        int code = idx_sh[wave * 32 + m];
        size_t gi = (size_t)(row0 + m) * D_DIM + d;
        float q  = w[d * K_DIM + code];
        float xv = x[gi];
        q_out[gi] = q;            // quantized_st == quantized numerically
        float diff = q - xv;
        lsum += diff * diff;
    }
    // Deterministic per-wave partial (fixed shuffle tree, then store).
    #pragma unroll
    for (int off = 16; off >= 1; off >>= 1)
        lsum += __shfl_xor(lsum, off, 32);
    if (lane == 0)
        loss_parts[blockIdx.x * WAVES_PER_BLOCK + wave] = lsum;
}

// ---------------------------------------------------------------------------
// Kernel 3: fixed-order reductions -> loss & perplexity scalars.
// ---------------------------------------------------------------------------
__global__ void __launch_bounds__(256)
vq_final_kernel(const unsigned int* __restrict__ counts,
                const float* __restrict__ loss_parts,
                float* __restrict__ out_scalars) {
    __shared__ float sh_ent[256];
    __shared__ float sh_loss[256];

    float ent = 0.0f, ls = 0.0f;
    for (int k = threadIdx.x; k < K_DIM; k += 256) {
        float p = (float)counts[k] * (1.0f / (float)N_ROWS);
        ent += p * logf(p + 1e-10f);
    }
    for (int k = threadIdx.x; k < N_LOSS_PARTS; k += 256)
        ls += loss_parts[k];

    sh_ent[threadIdx.x]  = ent;
    sh_loss[threadIdx.x] = ls;
    __syncthreads();
    for (int off = 128; off >= 1; off >>= 1) {
        if (threadIdx.x < off) {
            sh_ent[threadIdx.x]  += sh_ent[threadIdx.x + off];
            sh_loss[threadIdx.x] += sh_loss[threadIdx.x + off];
        }
        __syncthreads();
    }
    if (threadIdx.x == 0) {
        // loss = (q_latent + 0.25*e_latent)*1.0 = 1.25 * mean((q-x)^2)
        out_scalars[0] = 1.25f * sh_loss[0] / (float)(N_ROWS * D_DIM);
        out_scalars[1] = expf(-sh_ent[0]);
    }
}

// ---------------------------------------------------------------------------
extern "C" void kernel_launch(void* const* d_in, const int* in_sizes, int n_in,
                              void* d_out, int out_size, void* d_ws, size_t ws_size,
                              hipStream_t stream) {
    const float* x = (const float*)d_in[0];   // [32*32*32, 64] flat
    const float* w = (const float*)d_in[1];   // [64, 2048]
    float* out = (float*)d_out;               // 2097152 quantized + loss + perplexity

    // Workspace layout
    float*        wsq        = (float*)d_ws;                       // 2048 f32
    unsigned int* counts     = (unsigned int*)(wsq + K_DIM);       // 2048 u32
    float*        loss_parts = (float*)(counts + K_DIM);           // 1024 f32

    vq_init_kernel<<<K_DIM / 256, 256, 0, stream>>>(w, wsq, counts);

    vq_main_kernel<<<MAIN_BLOCKS, 256, 0, stream>>>(
        x, w, wsq, out, counts, loss_parts);

    vq_final_kernel<<<1, 256, 0, stream>>>(
        counts, loss_parts, out + (size_t)N_ROWS * D_DIM);
}